// BertNERCRFModel_84207128805567
// MI455X (gfx1250) — compile-verified
//
#include <hip/hip_runtime.h>

typedef __attribute__((ext_vector_type(2))) float v2f;
typedef __attribute__((ext_vector_type(8))) float v8f;

#define Bb 64
#define Tt 512
#define Hh 1024
#define Cc 21
#define NPAD 32
#define KC 256          // K chunk staged in LDS
#define KP (KC + 2)     // padded stride (floats) to avoid LDS bank conflicts

// ---------------------------------------------------------------------------
// Kernel 1: logits = seq_out @ W + bias  via V_WMMA_F32_16X16X4_F32
//   grid: (B*T)/128 workgroups of 256 threads (8 waves)
//   each wave: 16-row M tile x 32 N (two v8f accumulators), K loop over 1024
// ---------------------------------------------------------------------------
__global__ __launch_bounds__(256) void gemm_logits_kernel(
    const float* __restrict__ A,      // [B*T, H]
    const float* __restrict__ W,      // [H, C]
    const float* __restrict__ bias,   // [C]
    float* __restrict__ logits)       // [B*T, C]
{
    __shared__ float wlds[NPAD * KP]; // W chunk transposed: [n][k], padded

    const int tid  = threadIdx.x;
    const int lane = tid & 31;
    const int wave = tid >> 5;
    const int half = lane >> 4;       // 0: lanes 0-15, 1: lanes 16-31
    const int l16  = lane & 15;
    const long mbase = (long)blockIdx.x * 128 + (long)wave * 16;

    // C matrix seeded with bias (broadcast down rows)
    float bv0 = (l16 < Cc)        ? bias[l16]      : 0.0f;
    float bv1 = ((16 + l16) < Cc) ? bias[16 + l16] : 0.0f;
    v8f acc0, acc1;
    for (int r = 0; r < 8; ++r) { acc0[r] = bv0; acc1[r] = bv1; }

    const float* aptr = A + (mbase + l16) * (long)Hh; // this lane's A row

    for (int kc = 0; kc < Hh; kc += KC) {
        // Stage W[kc..kc+KC) transposed+padded into LDS: wlds[n*KP + klocal]
        __syncthreads();
        for (int n = 0; n < NPAD; ++n) {
            float v = 0.0f;
            if (n < Cc) v = W[(long)(kc + tid) * Cc + n];
            wlds[n * KP + tid] = v;
        }
        __syncthreads();

        // prefetch next A chunk while computing this one
        if (kc + KC < Hh) __builtin_prefetch(aptr + (kc + KC), 0, 0);

        for (int k4 = 0; k4 < KC; k4 += 4) {
            const int kk = k4 + 2 * half;
            // A fragment (16x4 f32): lane holds row mbase+l16, K = kk, kk+1
            v2f a = *(const v2f*)(aptr + kc + kk);
            // B fragments (4x16 f32) from LDS: lane holds N, K = kk, kk+1
            v2f b0 = *(const v2f*)(&wlds[l16 * KP + kk]);
            v2f b1 = *(const v2f*)(&wlds[(16 + l16) * KP + kk]);
            acc0 = __builtin_amdgcn_wmma_f32_16x16x4_f32(
                false, a, false, b0, (short)0, acc0, false, false);
            acc1 = __builtin_amdgcn_wmma_f32_16x16x4_f32(
                false, a, false, b1, (short)0, acc1, false, false);
        }
    }

    // C/D layout: VGPR r -> row (mbase + r + 8*half), col l16 (acc0) / 16+l16 (acc1)
    for (int r = 0; r < 8; ++r) {
        long row = mbase + r + 8 * half;
        logits[row * Cc + l16] = acc0[r];
        if (16 + l16 < Cc) logits[row * Cc + 16 + l16] = acc1[r];
    }
}

// ---------------------------------------------------------------------------
// Kernel 2: Viterbi forward + backward + one-hot. One wave32 per batch.
//   lane = current class (lanes 0..20 active for class work)
// ---------------------------------------------------------------------------
__global__ __launch_bounds__(32) void crf_kernel(
    const float* __restrict__ logits,  // [B,T,C]
    const int*   __restrict__ mask,    // [B,T]
    const float* __restrict__ trans,   // [C,C]
    int*         __restrict__ bp_ws,   // [B, T-1, C]
    float*       __restrict__ one_hot) // [B,T,C]
{
    const int b    = blockIdx.x;
    const int lane = threadIdx.x;
    const float NEG = -3.0e38f;

    // seq_len = sum(mask[b,:] == 1), wave reduction
    int s = 0;
    for (int t = lane; t < Tt; t += 32) s += (mask[b * Tt + t] == 1) ? 1 : 0;
    for (int off = 16; off > 0; off >>= 1) s += __shfl_xor(s, off, 32);
    const int seq_len = s;

    // transitions column for this lane's class: tcol[prev] = trans[prev][lane]
    float tcol[Cc];
    for (int p = 0; p < Cc; ++p)
        tcol[p] = (lane < Cc) ? trans[p * Cc + lane] : NEG;

    const float* lg = logits + (long)b * Tt * Cc;
    int* bpb = bp_ws + (long)b * (Tt - 1) * Cc;

    float state = (lane < Cc) ? lg[lane] : NEG;

    for (int t = 1; t < Tt; ++t) {
        float best = NEG; int bestp = 0;
        for (int p = 0; p < Cc; ++p) {
            float sp = __shfl(state, p, 32) + tcol[p];  // state[prev] broadcast
            if (sp > best) { best = sp; bestp = p; }    // first-max tie-break
        }
        const bool valid = (t < seq_len);
        float lt = (lane < Cc) ? lg[t * Cc + lane] : NEG;
        state = valid ? (best + lt) : state;
        int bpv = valid ? bestp : lane;                 // identity when masked
        if (lane < Cc) bpb[(t - 1) * Cc + lane] = bpv;
    }

    // last_tag = argmax(state) with lowest-index tie-break
    float v = (lane < Cc) ? state : NEG;
    int idx = lane;
    for (int off = 16; off > 0; off >>= 1) {
        float v2 = __shfl_xor(v, off, 32);
        int   i2 = __shfl_xor(idx, off, 32);
        if (v2 > v || (v2 == v && i2 < idx)) { v = v2; idx = i2; }
    }
    const int last_tag = __shfl(idx, 0, 32);

    // zero this batch's one-hot slice, then backtrace and scatter 1.0
    float* oh = one_hot + (long)b * Tt * Cc;
    for (int i = lane; i < Tt * Cc; i += 32) oh[i] = 0.0f;

    if (lane == 0) {
        int cur = last_tag;
        oh[(Tt - 1) * Cc + cur] = 1.0f;
        for (int t = Tt - 2; t >= 0; --t) {
            cur = bpb[t * Cc + cur];
            oh[t * Cc + cur] = 1.0f;
        }
    }
}

// ---------------------------------------------------------------------------
extern "C" void kernel_launch(void* const* d_in, const int* in_sizes, int n_in,
                              void* d_out, int out_size, void* d_ws, size_t ws_size,
                              hipStream_t stream) {
    const float* seq   = (const float*)d_in[0];  // [B,T,H] f32
    const int*   mask  = (const int*)  d_in[1];  // [B,T]   i32
    const float* W     = (const float*)d_in[2];  // [H,C]   f32
    const float* bias  = (const float*)d_in[3];  // [C]     f32
    const float* trans = (const float*)d_in[4];  // [C,C]   f32

    float* one_hot = (float*)d_out;                          // output 0
    float* logits  = (float*)d_out + (long)Bb * Tt * Cc;     // output 1
    int*   bp      = (int*)d_ws;                             // [B, T-1, C]

    gemm_logits_kernel<<<dim3((Bb * Tt) / 128), dim3(256), 0, stream>>>(
        seq, W, bias, logits);
    crf_kernel<<<dim3(Bb), dim3(32), 0, stream>>>(
        logits, mask, trans, bp, one_hot);
}